// CropAndResize3D_89481348645400
// MI455X (gfx1250) — compile-verified
//
#include <hip/hip_runtime.h>

#define NBOX 256
#define NCH  32
#define LDIM 64
#define CROP 16
#define TPAD 264   // LDS row pitch (floats); 2*264 % 64 == 16 -> lane-half reads hit disjoint banks

typedef __attribute__((ext_vector_type(2))) float v2f;
typedef __attribute__((ext_vector_type(8))) float v8f;

__global__ __launch_bounds__(256)
void crop_resize3d_wmma_kernel(const float* __restrict__ img,
                               const float* __restrict__ boxes,
                               const int*   __restrict__ box_ind,
                               float*       __restrict__ out)
{
    __shared__ float sT[32][TPAD];   // T[z-slot][p], p = yo*16+xo
    __shared__ int   s_i0[3][16];    // axis 0=z,1=y,2=x
    __shared__ int   s_i1[3][16];
    __shared__ float s_fr[3][16];
    __shared__ int   s_vl[3][16];

    const int tid = threadIdx.x;
    const int c   = blockIdx.x & (NCH - 1);
    const int b   = blockIdx.x >> 5;

    // ---- Per-box interpolation tables (reference _coords / _lerp_idx) ----
    if (tid < 48) {
        const int a = tid >> 4;          // 0=z, 1=y, 2=x
        const int i = tid & 15;
        // boxes columns: [x1,y1,z1,x2,y2,z2] -> z:(2,5) y:(1,4) x:(0,3)
        const int col_lo = (a == 0) ? 2 : (a == 1) ? 1 : 0;
        const float lo = boxes[b * 6 + col_lo];
        const float hi = boxes[b * 6 + col_lo + 3];
        const float step  = (hi - lo) * 63.0f / 15.0f;
        const float coord = lo * 63.0f + (float)i * step;
        const int valid = (coord >= 0.0f) && (coord <= 63.0f);
        const float cc = fminf(fmaxf(coord, 0.0f), 63.0f);
        const float fl = floorf(cc);
        const int i0 = (int)fl;
        s_i0[a][i] = i0;
        s_i1[a][i] = min(i0 + 1, LDIM - 1);
        s_fr[a][i] = cc - fl;
        s_vl[a][i] = valid;
    }
    __syncthreads();

    // ---- Stage 1: bilinear (y,x) gather-interp at 32 z-slots -> LDS ----
    const int   xo = tid & 15;
    const int   yo = (tid >> 4) & 15;
    const int   x0 = s_i0[2][xo], x1 = s_i1[2][xo];
    const float fx = s_fr[2][xo];
    const int   y0 = s_i0[1][yo], y1 = s_i1[1][yo];
    const float fy = s_fr[1][yo];

    const long  n    = box_ind[b];
    const float* base = img + (((long)n * NCH + c) * LDIM) * (LDIM * LDIM);

    #pragma unroll 4
    for (int zs = 0; zs < 32; ++zs) {
        const int zo = zs >> 1;
        const int zi = (zs & 1) ? s_i1[0][zo] : s_i0[0][zo];
        const float* pz = base + (long)zi * (LDIM * LDIM);
        const float* r0 = pz + y0 * LDIM;
        const float* r1 = pz + y1 * LDIM;
        const float v00 = r0[x0], v01 = r0[x1];
        const float v10 = r1[x0], v11 = r1[x1];
        const float cx0 = v00 * (1.0f - fx) + v01 * fx;
        const float cx1 = v10 * (1.0f - fx) + v11 * fx;
        sT[zs][tid] = cx0 * (1.0f - fy) + cx1 * fy;
    }
    __syncthreads();

    // ---- Stage 2: z-combine as fp32 WMMA: out[16 zo x 256 p] = Wz(16x32) * T(32x256)
    const int lane = tid & 31;       // wave32
    const int wave = tid >> 5;       // 0..7 -> each wave owns 2 N-tiles (tile index == yo)
    const int nlo  = lane & 15;
    const int koff = (lane >> 4) << 1;       // 0 for lanes 0-15, 2 for lanes 16-31
    const float fzm  = s_fr[0][nlo];         // A-row m == nlo
    const int   slot = 2 * nlo;              // Wz[m][2m]=(1-fz), Wz[m][2m+1]=fz

    const long obase = (((long)b * NCH + c) * CROP) * 256;   // out[b][c][zo][p]

    for (int t = wave * 2; t < wave * 2 + 2; ++t) {
        v8f acc = {};
        #pragma unroll
        for (int kk = 0; kk < 8; ++kk) {
            const int kb = kk * 4 + koff;    // this lane-half's K pair base
            v2f a;                            // A 16x4 layout (ISA 7.12.2)
            a.x = (kb + 0 == slot) ? (1.0f - fzm) : ((kb + 0 == slot + 1) ? fzm : 0.0f);
            a.y = (kb + 1 == slot) ? (1.0f - fzm) : ((kb + 1 == slot + 1) ? fzm : 0.0f);
            v2f bm;                           // B 4x16 layout (mirror of A)
            bm.x = sT[kb + 0][t * 16 + nlo];
            bm.y = sT[kb + 1][t * 16 + nlo];
            acc = __builtin_amdgcn_wmma_f32_16x16x4_f32(
                      /*neg_a=*/false, a, /*neg_b=*/false, bm,
                      /*c_mod=*/(short)0, acc, /*reuse_a=*/false, /*reuse_b=*/false);
        }
        // D layout: VGPR r -> M=r (lanes 0-15) / M=r+8 (lanes 16-31), N=lane&15
        const int mbase = (lane >> 4) * 8;
        const int vyx   = s_vl[1][t] & s_vl[2][nlo];
        #pragma unroll
        for (int r = 0; r < 8; ++r) {
            const int zo = mbase + r;
            const float val = (s_vl[0][zo] & vyx) ? acc[r] : 0.0f;
            out[obase + (long)zo * 256 + t * 16 + nlo] = val;
        }
    }
}

extern "C" void kernel_launch(void* const* d_in, const int* in_sizes, int n_in,
                              void* d_out, int out_size, void* d_ws, size_t ws_size,
                              hipStream_t stream) {
    (void)in_sizes; (void)n_in; (void)out_size; (void)d_ws; (void)ws_size;
    const float* img     = (const float*)d_in[0];
    const float* boxes   = (const float*)d_in[1];
    const int*   box_ind = (const int*)d_in[2];
    float*       out     = (float*)d_out;

    crop_resize3d_wmma_kernel<<<dim3(NBOX * NCH), dim3(256), 0, stream>>>(
        img, boxes, box_ind, out);
}